// Model_68985764708842
// MI455X (gfx1250) — compile-verified
//
#include <hip/hip_runtime.h>
#include <stdint.h>

// Depthwise causal conv1d, B=8, T=4096, D=1024, K=4  (f32 in / f32 out)
//
// y[b,t,d] = sum_{k=0..3} W[d,k] * x[b, t-3+k, d]   (zero-padded on the left)
//
// Memory-bound: ~268 MB min traffic -> ~11.5 us floor at 23.3 TB/s.
// Strategy: stream x through LDS with double-buffered async copies
// (GLOBAL_LOAD_ASYNC_TO_LDS_B128, tracked with ASYNCcnt), keep the 3-deep
// time history in registers so every x element is fetched from HBM once.
// Output stores are non-temporal (write-once stream).

#define B_ 8
#define T_ 4096
#define D_ 1024
#define K_ 4

#define TILE_T   32                 // timesteps per block -> 1024 blocks total
#define CHUNK_T  4                  // timesteps per async chunk (4 rows * 4KB = 16KB)
#define NCHUNK   (TILE_T / CHUNK_T) // 8
#define THREADS  256                // D_/4 channels -> 4 channels (one v4f) per thread

typedef __attribute__((ext_vector_type(4))) float v4f;   // native vector -> b128 + NT builtin OK

__device__ __forceinline__ void async_copy_b128(uint32_t lds_byte_addr, const void* gaddr) {
    // per-lane: copies 16 bytes from the global address in vaddr to the LDS address in vdst, tracked with ASYNCcnt
    asm volatile("global_load_async_to_lds_b128 %0, %1, off"
                 :: "v"(lds_byte_addr), "v"((uint64_t)(uintptr_t)gaddr)
                 : "memory");
}

__device__ __forceinline__ void issue_chunk(const float* __restrict__ xb,
                                            int t_base, float (*buf)[D_], int tid) {
    // one chunk = CHUNK_T rows of 4KB; each of the 256 threads copies 16B per row
    // -> each wave32 issues exactly CHUNK_T async ops per chunk (ASYNCcnt += 4)
    #pragma unroll
    for (int r = 0; r < CHUNK_T; ++r) {
        uint32_t lds_off = (uint32_t)(uintptr_t)(const void*)&buf[r][tid * 4];
        const void* g = xb + (size_t)(t_base + r) * D_ + tid * 4;
        async_copy_b128(lds_off, g);
    }
}

__global__ __launch_bounds__(THREADS)
void dwconv_causal_k4(const float* __restrict__ x,
                      const float* __restrict__ W,
                      float* __restrict__ y) {
    __shared__ float lds[2][CHUNK_T][D_];   // 32 KB double buffer

    const int tid  = threadIdx.x;
    const int b    = blockIdx.x / (T_ / TILE_T);
    const int tile = blockIdx.x % (T_ / TILE_T);
    const int t0   = tile * TILE_T;
    const int d0   = tid * 4;

    const float* xb = x + (size_t)b * T_ * D_;
    float*       yb = y + (size_t)b * T_ * D_;

    // ---- weights: W is [D,4] row-major; transpose 4 channels x 4 taps into
    //      per-tap vectors (w0 pairs with x[t-3], w3 with x[t]).
    v4f c0 = *(const v4f*)(W + (size_t)(d0 + 0) * K_);
    v4f c1 = *(const v4f*)(W + (size_t)(d0 + 1) * K_);
    v4f c2 = *(const v4f*)(W + (size_t)(d0 + 2) * K_);
    v4f c3 = *(const v4f*)(W + (size_t)(d0 + 3) * K_);
    v4f w0 = (v4f){c0.x, c1.x, c2.x, c3.x};
    v4f w1 = (v4f){c0.y, c1.y, c2.y, c3.y};
    v4f w2 = (v4f){c0.z, c1.z, c2.z, c3.z};
    v4f w3 = (v4f){c0.w, c1.w, c2.w, c3.w};

    // ---- register sliding-window history: x[t0-3], x[t0-2], x[t0-1]
    v4f h0 = (v4f){0.f, 0.f, 0.f, 0.f}, h1 = h0, h2 = h0;
    if (t0 >= 3) {  // tile 0 starts at t=0 -> zero history; all other tiles t0>=32
        h0 = *(const v4f*)(xb + (size_t)(t0 - 3) * D_ + d0);
        h1 = *(const v4f*)(xb + (size_t)(t0 - 2) * D_ + d0);
        h2 = *(const v4f*)(xb + (size_t)(t0 - 1) * D_ + d0);
    }

    // ---- prologue: put two chunks in flight
    issue_chunk(xb, t0 + 0 * CHUNK_T, lds[0], tid);
    issue_chunk(xb, t0 + 1 * CHUNK_T, lds[1], tid);

    for (int c = 0; c < NCHUNK; ++c) {
        // chunk c is complete once at most one chunk (4 async ops) is in flight
        if (c + 1 < NCHUNK) {
            asm volatile("s_wait_asynccnt 4" ::: "memory");
        } else {
            asm volatile("s_wait_asynccnt 0" ::: "memory");
        }
        __syncthreads();   // make all waves' async-written LDS visible

        float (*buf)[D_] = lds[c & 1];
        const int tbase = t0 + c * CHUNK_T;
        #pragma unroll
        for (int r = 0; r < CHUNK_T; ++r) {
            v4f xr = *(const v4f*)(&buf[r][d0]);   // ds_load_b128
            v4f o;
            o.x = fmaf(w3.x, xr.x, fmaf(w2.x, h2.x, fmaf(w1.x, h1.x, w0.x * h0.x)));
            o.y = fmaf(w3.y, xr.y, fmaf(w2.y, h2.y, fmaf(w1.y, h1.y, w0.y * h0.y)));
            o.z = fmaf(w3.z, xr.z, fmaf(w2.z, h2.z, fmaf(w1.z, h1.z, w0.z * h0.z)));
            o.w = fmaf(w3.w, xr.w, fmaf(w2.w, h2.w, fmaf(w1.w, h1.w, w0.w * h0.w)));
            h0 = h1; h1 = h2; h2 = xr;
            // write-once output stream: non-temporal global_store_b128
            __builtin_nontemporal_store(o, (v4f*)(yb + (size_t)(tbase + r) * D_ + d0));
        }

        __syncthreads();   // everyone done reading buf before it is overwritten
        if (c + 2 < NCHUNK)
            issue_chunk(xb, t0 + (c + 2) * CHUNK_T, lds[c & 1], tid);
    }
}

extern "C" void kernel_launch(void* const* d_in, const int* in_sizes, int n_in,
                              void* d_out, int out_size, void* d_ws, size_t ws_size,
                              hipStream_t stream) {
    const float* x = (const float*)d_in[0];   // [B,T,D] f32
    const float* W = (const float*)d_in[1];   // [D,K]   f32
    float* y = (float*)d_out;                 // [B,T,D] f32

    dim3 grid(B_ * (T_ / TILE_T));            // 1024 blocks
    dim3 block(THREADS);                      // 256 threads = 8 wave32
    dwconv_causal_k4<<<grid, block, 0, stream>>>(x, W, y);
}